// DetectionLayer_78013785965183
// MI455X (gfx1250) — compile-verified
//
#include <hip/hip_runtime.h>
#include <stdint.h>

// YOLO detection layer for MI455X (gfx1250).
// Bandwidth-bound permute+activation (~377 MB traffic -> ~16us floor @23.3TB/s).
// Phase 1: TDM (tensor_load_to_lds) DMAs an 85 x nS tile into LDS with +1-dword
//          row padding (conflict-free reads). Fallback: async global->LDS b32.
// Phase 2: activation + fully coalesced contiguous b32 stores.

#define G      76
#define GG     5776          // 76*76
#define NCH    85            // 5 + 80 classes
#define NA     3
#define NB     32
#define STRIDE 8.0f          // 608 / 76
#define TS     128           // spatial tile (power of two)
#define NTILES 46            // ceil(5776 / 128), last tile has 16
#define BLK    256           // 8 wave32 waves

#if defined(__has_builtin)
#  if __has_builtin(__builtin_amdgcn_tensor_load_to_lds)
#    define USE_TDM 1
#  endif
#  if __has_builtin(__builtin_amdgcn_global_load_async_to_lds_b32)
#    define USE_ASYNC_LDS 1
#  endif
#endif
#ifndef USE_TDM
#  define USE_TDM 0
#endif
#ifndef USE_ASYNC_LDS
#  define USE_ASYNC_LDS 0
#endif

typedef __attribute__((address_space(1))) int as1_int;   // global
typedef __attribute__((address_space(3))) int as3_int;   // LDS
typedef unsigned int v4u __attribute__((ext_vector_type(4)));
typedef int          v4i __attribute__((ext_vector_type(4)));
typedef int          v8i __attribute__((ext_vector_type(8)));

__global__ __launch_bounds__(BLK) void yolo_detect_kernel(
    const float* __restrict__ x, const float* __restrict__ anchors,
    float* __restrict__ out)
{
  // TDM layout: [c][s], row stride 129 dwords (padded). Async layout: [s][c].
  __shared__ float tile[NCH * (TS + 1) + 8];

  const int tileIdx = blockIdx.x;          // 0..NTILES-1
  const int plane   = blockIdx.y;          // 0..95  (= b*3 + a)
  const int b   = plane / NA;
  const int a   = plane - b * NA;
  const int tid = threadIdx.x;

  const int tile_start = tileIdx * TS;     // flat spatial offset within plane
  const int nS = min(TS, GG - tile_start);

  // input:  x[b][a*85 + c][spatial]   (plane-contiguous per channel)
  const unsigned in_base  = (unsigned)(b * (NA * NCH) + a * NCH) * GG
                          + (unsigned)tile_start;
  // output: out[b][a*GG + spatial][c] -> one contiguous chunk of nS*85 floats
  const unsigned out_base = ((unsigned)(b * NA + a) * GG + (unsigned)tile_start)
                          * (unsigned)NCH;

  const float aw = anchors[2 * a + 0];     // stride scaling cancels exactly
  const float ah = anchors[2 * a + 1];

#if USE_TDM
  // ---- Phase 1 (TDM): one DMA moves the whole 85 x nS tile -------------
  const int RS = (nS == TS) ? (TS + 1) : nS;     // LDS row stride in dwords
  if (tid < 32) {                                 // wave 0 only (TDM ignores EXEC)
    const unsigned lds_off =
        (unsigned)(unsigned long long)(as3_int*)(int*)(void*)&tile[0];
    const unsigned long long ga =
        (unsigned long long)(const void*)(x + in_base);

    v4u g0;
    g0[0] = 1u;                                   // count=1, user descriptor
    g0[1] = lds_off;                              // lds_addr (bytes)
    g0[2] = (unsigned)ga;                         // global_addr[31:0]
    g0[3] = ((unsigned)(ga >> 32) & 0x01FFFFFFu)  // global_addr[56:32]
          | 0x80000000u;                          // type=2 ("image")

    const unsigned padded = (nS == TS) ? 1u : 0u; // +1 dword per 128-dword row
    v8i g1;
    g1[0] = (int)((2u << 16)                      // data_size = 4 bytes
                | (padded << 20)                  // pad_enable
                | (padded ? (6u << 22) : 0u)      // pad_interval: 128 dwords
                | (0u << 25));                    // pad_amount: 1 dword
    g1[1] = (int)((unsigned)nS  << 16);           // tensor_dim0[15:0]
    g1[2] = (int)((unsigned)NCH << 16);           // tensor_dim1[15:0]
    g1[3] = (int)((unsigned)nS  << 16);           // tile_dim0
    g1[4] = (int)(unsigned)NCH;                   // tile_dim1 (tile_dim2=0)
    g1[5] = (int)(unsigned)GG;                    // tensor_dim0_stride lo32
    g1[6] = 0;                                    // stride hi / dim1_stride lo
    g1[7] = 0;
    v4i g2 = {0, 0, 0, 0};                        // 2D tensor: groups 2/3 unused
    v4i g3 = {0, 0, 0, 0};
    v8i g4 = {0, 0, 0, 0, 0, 0, 0, 0};            // extra arg of 6-arg builtin

    __builtin_amdgcn_tensor_load_to_lds(g0, g1, g2, g3, g4, /*cpol*/0);
#  if __has_builtin(__builtin_amdgcn_s_wait_tensorcnt)
    __builtin_amdgcn_s_wait_tensorcnt(0);
#  else
    asm volatile("s_wait_tensorcnt 0x0" ::: "memory");
#  endif
  }
  __syncthreads();
#else
  // ---- Phase 1 (fallback): per-lane staging, transposed [s][c] ---------
  for (int e = tid; e < NCH * TS; e += BLK) {
    const int c = e >> 7;                  // TS == 128
    const int s = e & (TS - 1);
    if (s < nS) {
      const unsigned gidx = in_base + (unsigned)c * GG + (unsigned)s;
      const int      lidx = s * NCH + c;
#  if USE_ASYNC_LDS
      __builtin_amdgcn_global_load_async_to_lds_b32(
          (as1_int*)(int*)(void*)(x + gidx),
          (as3_int*)(int*)(void*)(&tile[lidx]), 0, 0);
#  else
      tile[lidx] = x[gidx];
#  endif
    }
  }
#  if USE_ASYNC_LDS
#    if __has_builtin(__builtin_amdgcn_s_wait_asynccnt)
  __builtin_amdgcn_s_wait_asynccnt(0);
#    else
  asm volatile("s_wait_asynccnt 0" ::: "memory");
#    endif
#  endif
  __syncthreads();
#endif

  // ---- Phase 2: activation + fully coalesced contiguous store ---------
  // o = s*85 + c. Since BLK = 256 == 3*85 + 1, (s,c) advance by a constant
  // increment each iteration: no per-element division needed.
  const int total = nS * NCH;
  int s = tid / NCH;                       // tid < 256 -> one cheap division
  int c = tid - s * NCH;
  for (int o = tid; o < total; o += BLK) {
#if USE_TDM
    const float v = tile[c * RS + s];      // lane stride RS=129 -> no conflicts
#else
    const float v = tile[o];               // linear, conflict-free
#endif
    float r;
    if (c == 2) {                          // bw = exp(w) * anchor_w
      r = __expf(v) * aw;
    } else if (c == 3) {                   // bh = exp(h) * anchor_h
      r = __expf(v) * ah;
    } else {
      const float sg = 1.0f / (1.0f + __expf(-v));
      if (c == 0) {                        // bx = (sig + grid_x) * stride
        const int sp = tile_start + s;
        r = (sg + (float)(sp % G)) * STRIDE;
      } else if (c == 1) {                 // by = (sig + grid_y) * stride
        const int sp = tile_start + s;
        r = (sg + (float)(sp / G)) * STRIDE;
      } else {                             // conf / classes: sigmoid
        r = sg;
      }
    }
    out[out_base + (unsigned)o] = r;       // contiguous b32 stores

    c += 1; s += 3;                        // o += 256  ==  (s+=3, c+=1) mod 85
    if (c >= NCH) { c -= NCH; s += 1; }
  }
}

extern "C" void kernel_launch(void* const* d_in, const int* in_sizes, int n_in,
                              void* d_out, int out_size, void* d_ws, size_t ws_size,
                              hipStream_t stream) {
  (void)in_sizes; (void)n_in; (void)out_size; (void)d_ws; (void)ws_size;
  const float* x       = (const float*)d_in[0];
  const float* anchors = (const float*)d_in[1];
  float*       out     = (float*)d_out;

  dim3 grid(NTILES, NB * NA);   // 46 spatial tiles x 96 (batch, anchor) planes
  yolo_detect_kernel<<<grid, BLK, 0, stream>>>(x, anchors, out);
}